// PSREG_features_list_45234595561877
// MI455X (gfx1250) — compile-verified
//
#include <hip/hip_runtime.h>
#include <math.h>

typedef float v2f __attribute__((ext_vector_type(2)));
typedef float v8f __attribute__((ext_vector_type(8)));

#define GAMMA_C 0.005f
#define EPS_C   1e-5f
static constexpr int MV = 4;
static constexpr int KC = 256;
static constexpr int FC = 64;

// workspace layout (float offsets)
static constexpr int WS_Q    = 0;      // [256]
static constexpr int WS_BETA = 256;    // [1]
static constexpr int WS_L    = 512;    // [4*256]
static constexpr int WS_W    = 1536;   // [4*3*256]
static constexpr int WS_S2   = 4608;   // [4*256]
static constexpr int WS_WF   = 5632;   // [256*64]
static constexpr int WS_THT  = 22016;  // [64*256]  (theta transposed, [F][K])
static constexpr int WS_ACD  = 38400;  // [4*N]     (per-point a-denominators)

// ---------------------------------------------------------------- init
__global__ __launch_bounds__(256) void kInit(
    const float* __restrict__ X0, const float* __restrict__ Q0,
    const float* __restrict__ th0,
    float* __restrict__ Xg, float* __restrict__ Qw,
    float* __restrict__ betaW, float* __restrict__ thT)
{
  __shared__ float red[256];
  const int gid = blockIdx.x * 256 + threadIdx.x;
  if (gid < KC * FC) {
    int k = gid / FC, f = gid % FC;
    thT[f * KC + k] = th0[gid];
  }
  if (gid < KC * 3) Xg[gid] = X0[gid];
  if (gid < KC)     Qw[gid] = Q0[gid];
  if (blockIdx.x == 0) {
    float v = (threadIdx.x < KC) ? Q0[threadIdx.x] : 0.f;
    red[threadIdx.x] = v;
    __syncthreads();
    for (int s = 128; s > 0; s >>= 1) {
      if (threadIdx.x < s) red[threadIdx.x] += red[threadIdx.x + s];
      __syncthreads();
    }
    if (threadIdx.x == 0) {
      float mq = red[0] / (float)KC;
      betaW[0] = GAMMA_C * mq * sqrtf(mq);   // gamma * mean(Q)^1.5
    }
  }
}

__global__ __launch_bounds__(256) void kMean(
    const float* __restrict__ Vs, float* __restrict__ Rg, float* __restrict__ tg, int N)
{
  __shared__ float red[256];
  const int m = blockIdx.x / 3, d = blockIdx.x % 3;
  float s = 0.f;
  for (int n = threadIdx.x; n < N; n += 256) s += Vs[((size_t)m * 3 + d) * N + n];
  red[threadIdx.x] = s;
  __syncthreads();
  for (int st = 128; st > 0; st >>= 1) {
    if (threadIdx.x < st) red[threadIdx.x] += red[threadIdx.x + st];
    __syncthreads();
  }
  if (threadIdx.x == 0) tg[m * 3 + d] = -red[0] / (float)N;
  if (d == 0 && threadIdx.x < 9)
    Rg[m * 9 + threadIdx.x] = ((threadIdx.x % 4) == 0) ? 1.f : 0.f;  // identity
}

__global__ __launch_bounds__(256) void kZero(float* __restrict__ ws)
{
  const int i = 512 + blockIdx.x * 256 + threadIdx.x;
  if (i < 22016) ws[i] = 0.f;
}

__global__ __launch_bounds__(256) void kTV(
    const float* __restrict__ Vs, const float* __restrict__ Rg,
    const float* __restrict__ tg, float* __restrict__ TV, int N)
{
  const int m = blockIdx.y;
  const int n = blockIdx.x * 256 + threadIdx.x;
  if (n >= N) return;
  const float v0 = Vs[((size_t)m * 3 + 0) * N + n];
  const float v1 = Vs[((size_t)m * 3 + 1) * N + n];
  const float v2 = Vs[((size_t)m * 3 + 2) * N + n];
  #pragma unroll
  for (int d = 0; d < 3; ++d) {
    TV[((size_t)m * 3 + d) * N + n] =
        Rg[m * 9 + d * 3 + 0] * v0 + Rg[m * 9 + d * 3 + 1] * v1 +
        Rg[m * 9 + d * 3 + 2] * v2 + tg[m * 3 + d];
  }
}

// ---------------------------------------------------------------- E-step
// Fused: pyz = feat @ theta^T via V_WMMA_F32_16X16X4_F32, ds/exp chain,
// normalization denominators, and L / Wraw / Sv2 partial reductions.
__global__ __launch_bounds__(128) void kE(
    const float* __restrict__ Vs, const float* __restrict__ feat,
    const float* __restrict__ TV, const float* __restrict__ X,
    const float* __restrict__ Q,  const float* __restrict__ thT,
    const float* __restrict__ betaP,
    float* __restrict__ Lg, float* __restrict__ Wg, float* __restrict__ S2g,
    float* __restrict__ acden, int N)
{
  const int tid  = threadIdx.x;
  const int w    = tid >> 5;
  const int lane = tid & 31;
  const int half = lane >> 4;
  const int l15  = lane & 15;
  const int m    = blockIdx.y;
  const int nbase = blockIdx.x * 64 + w * 16;   // 16 points per wave

  __shared__ float sX[KC * 3];
  __shared__ float sQ[KC];
  __shared__ float sQc[KC];
  __shared__ float sFeat[4][16][FC + 1];
  __shared__ float sTV[4][3][16];
  __shared__ float sTVn[4][16];
  __shared__ float sVs[4][3][16];
  __shared__ float sVn[4][16];
  __shared__ float sAc[4][16];
  __shared__ float pL[KC];
  __shared__ float pW[3][KC];
  __shared__ float pS[KC];

  const float beta = betaP[0];

  for (int i = tid; i < KC; i += 128) {
    float q = Q[i];
    sQ[i] = q; sQc[i] = q * sqrtf(q);
    sX[i * 3 + 0] = X[i * 3 + 0];
    sX[i * 3 + 1] = X[i * 3 + 1];
    sX[i * 3 + 2] = X[i * 3 + 2];
    pL[i] = 0.f; pW[0][i] = 0.f; pW[1][i] = 0.f; pW[2][i] = 0.f; pS[i] = 0.f;
  }
  for (int i = lane; i < 16 * FC; i += 32) {
    int r = i >> 6, f = i & 63;
    int n = nbase + r;
    sFeat[w][r][f] = (n < N) ? feat[((size_t)m * N + n) * FC + f] : 0.f;
  }
  if (lane < 16) {
    int n = nbase + lane; int nc = n < N ? n : N - 1;
    float a0 = TV[((size_t)m * 3 + 0) * N + nc];
    float a1 = TV[((size_t)m * 3 + 1) * N + nc];
    float a2 = TV[((size_t)m * 3 + 2) * N + nc];
    sTV[w][0][lane] = a0; sTV[w][1][lane] = a1; sTV[w][2][lane] = a2;
    sTVn[w][lane] = a0 * a0 + a1 * a1 + a2 * a2;
  } else {
    int p = lane - 16; int n = nbase + p; int nc = n < N ? n : N - 1;
    float a0 = Vs[((size_t)m * 3 + 0) * N + nc];
    float a1 = Vs[((size_t)m * 3 + 1) * N + nc];
    float a2 = Vs[((size_t)m * 3 + 2) * N + nc];
    sVs[w][0][p] = a0; sVs[w][1][p] = a1; sVs[w][2][p] = a2;
    sVn[w][p] = a0 * a0 + a1 * a1 + a2 * a2;
  }
  __syncthreads();

  // ---- pass A: ac_den = sum_k ap*pyz + beta -------------------------
  float asum[8];
  #pragma unroll
  for (int v = 0; v < 8; ++v) asum[v] = 0.f;

  for (int kt = 0; kt < 16; ++kt) {
    v8f c;
    #pragma unroll
    for (int v = 0; v < 8; ++v) c[v] = 0.f;
    const int kcol = kt * 16 + l15;
    #pragma unroll
    for (int ff = 0; ff < 16; ++ff) {
      const int kb = ff * 4 + half * 2;
      v2f a, b;
      a.x = sFeat[w][l15][kb];     a.y = sFeat[w][l15][kb + 1];
      b.x = thT[kb * KC + kcol];   b.y = thT[(kb + 1) * KC + kcol];
      c = __builtin_amdgcn_wmma_f32_16x16x4_f32(false, a, false, b, (short)0, c, false, false);
    }
    const float q = sQ[kcol], qc = sQc[kcol];
    const float x0 = sX[kcol * 3], x1 = sX[kcol * 3 + 1], x2 = sX[kcol * 3 + 2];
    const float xn = x0 * x0 + x1 * x1 + x2 * x2;
    #pragma unroll
    for (int v = 0; v < 8; ++v) {
      const int p = half * 8 + v;
      float ds = sTVn[w][p] + xn -
                 2.f * (sTV[w][0][p] * x0 + sTV[w][1][p] * x1 + sTV[w][2][p] * x2);
      float ap = __expf(-0.5f * ds * q) * qc;
      asum[v] += ap * c[v];
    }
  }
  #pragma unroll
  for (int off = 1; off < 16; off <<= 1) {
    #pragma unroll
    for (int v = 0; v < 8; ++v) asum[v] += __shfl_xor(asum[v], off, 32);
  }
  if (l15 == 0) {
    #pragma unroll
    for (int v = 0; v < 8; ++v) {
      int p = half * 8 + v;
      float ac = asum[v] + beta;
      sAc[w][p] = ac;
      int n = nbase + p;
      if (n < N) acden[(size_t)m * N + n] = ac;
    }
  }
  __syncthreads();

  // ---- pass B: a = ap*pyz/ac_den, accumulate L, Wraw, Sv2 -----------
  for (int kt = 0; kt < 16; ++kt) {
    v8f c;
    #pragma unroll
    for (int v = 0; v < 8; ++v) c[v] = 0.f;
    const int kcol = kt * 16 + l15;
    #pragma unroll
    for (int ff = 0; ff < 16; ++ff) {
      const int kb = ff * 4 + half * 2;
      v2f a, b;
      a.x = sFeat[w][l15][kb];     a.y = sFeat[w][l15][kb + 1];
      b.x = thT[kb * KC + kcol];   b.y = thT[(kb + 1) * KC + kcol];
      c = __builtin_amdgcn_wmma_f32_16x16x4_f32(false, a, false, b, (short)0, c, false, false);
    }
    const float q = sQ[kcol], qc = sQc[kcol];
    const float x0 = sX[kcol * 3], x1 = sX[kcol * 3 + 1], x2 = sX[kcol * 3 + 2];
    const float xn = x0 * x0 + x1 * x1 + x2 * x2;
    float sa = 0.f, sw0 = 0.f, sw1 = 0.f, sw2 = 0.f, ss = 0.f;
    #pragma unroll
    for (int v = 0; v < 8; ++v) {
      const int p = half * 8 + v;
      float ds = sTVn[w][p] + xn -
                 2.f * (sTV[w][0][p] * x0 + sTV[w][1][p] * x1 + sTV[w][2][p] * x2);
      float ap = __expf(-0.5f * ds * q) * qc;
      float av = ap * c[v] / sAc[w][p];
      sa  += av;
      sw0 += sVs[w][0][p] * av;
      sw1 += sVs[w][1][p] * av;
      sw2 += sVs[w][2][p] * av;
      ss  += sVn[w][p] * av;
    }
    atomicAdd(&pL[kcol], sa);
    atomicAdd(&pW[0][kcol], sw0);
    atomicAdd(&pW[1][kcol], sw1);
    atomicAdd(&pW[2][kcol], sw2);
    atomicAdd(&pS[kcol], ss);
  }
  __syncthreads();
  for (int i = tid; i < KC; i += 128) {
    atomicAdd(&Lg[m * KC + i], pL[i]);
    atomicAdd(&Wg[(m * 3 + 0) * KC + i], pW[0][i]);
    atomicAdd(&Wg[(m * 3 + 1) * KC + i], pW[1][i]);
    atomicAdd(&Wg[(m * 3 + 2) * KC + i], pW[2][i]);
    atomicAdd(&S2g[m * KC + i], pS[i]);
  }
}

// ---------------------------------------------------------------- wfeat
// w[k,f] += (ap/ac_den)^T @ features via WMMA; each wave covers 256 points.
__global__ __launch_bounds__(128) void kWF(
    const float* __restrict__ feat, const float* __restrict__ TV,
    const float* __restrict__ X, const float* __restrict__ Q,
    const float* __restrict__ acden, float* __restrict__ wfg, int N)
{
  const int tid  = threadIdx.x;
  const int w    = tid >> 5;
  const int lane = tid & 31;
  const int half = lane >> 4;
  const int l15  = lane & 15;
  const int m    = blockIdx.y;
  const int pbase = (blockIdx.x * 4 + w) * 256;

  __shared__ float sX[KC * 3];
  __shared__ float sQ[KC];
  __shared__ float sQc[KC];
  __shared__ float sLt[4][16][17];

  for (int i = tid; i < KC; i += 128) {
    float q = Q[i]; sQ[i] = q; sQc[i] = q * sqrtf(q);
    sX[i * 3 + 0] = X[i * 3 + 0];
    sX[i * 3 + 1] = X[i * 3 + 1];
    sX[i * 3 + 2] = X[i * 3 + 2];
  }
  __syncthreads();

  for (int kt = 0; kt < 16; ++kt) {
    v8f cf[4];
    #pragma unroll
    for (int ft = 0; ft < 4; ++ft) {
      #pragma unroll
      for (int v = 0; v < 8; ++v) cf[ft][v] = 0.f;
    }
    for (int pt = 0; pt < 16; ++pt) {
      const int n  = pbase + pt * 16 + l15;
      const bool ok = (n < N);
      const int nc = ok ? n : 0;
      const float t0 = TV[((size_t)m * 3 + 0) * N + nc];
      const float t1 = TV[((size_t)m * 3 + 1) * N + nc];
      const float t2 = TV[((size_t)m * 3 + 2) * N + nc];
      const float tn = t0 * t0 + t1 * t1 + t2 * t2;
      const float ad = acden[(size_t)m * N + nc];
      const float adi = ok ? 1.f / ad : 0.f;
      #pragma unroll
      for (int j = 0; j < 8; ++j) {
        int k = kt * 16 + half * 8 + j;
        float x0 = sX[k * 3], x1 = sX[k * 3 + 1], x2 = sX[k * 3 + 2];
        float ds = tn + x0 * x0 + x1 * x1 + x2 * x2 -
                   2.f * (t0 * x0 + t1 * x1 + t2 * x2);
        float ap = __expf(-0.5f * ds * sQ[k]) * sQc[k];
        sLt[w][half * 8 + j][l15] = ap * adi;
      }
      __syncthreads();
      #pragma unroll
      for (int pp = 0; pp < 4; ++pp) {
        v2f a;
        a.x = sLt[w][l15][pp * 4 + half * 2];
        a.y = sLt[w][l15][pp * 4 + half * 2 + 1];
        const int pr = pbase + pt * 16 + pp * 4 + half * 2;
        #pragma unroll
        for (int ft = 0; ft < 4; ++ft) {
          v2f b;
          const int f = ft * 16 + l15;
          b.x = (pr     < N) ? feat[((size_t)m * N + pr    ) * FC + f] : 0.f;
          b.y = (pr + 1 < N) ? feat[((size_t)m * N + pr + 1) * FC + f] : 0.f;
          cf[ft] = __builtin_amdgcn_wmma_f32_16x16x4_f32(false, a, false, b, (short)0, cf[ft], false, false);
        }
      }
      __syncthreads();
    }
    #pragma unroll
    for (int v = 0; v < 8; ++v) {
      const int kr = kt * 16 + half * 8 + v;
      atomicAdd(&wfg[kr * FC + 0 * 16 + l15], cf[0][v]);
      atomicAdd(&wfg[kr * FC + 1 * 16 + l15], cf[1][v]);
      atomicAdd(&wfg[kr * FC + 2 * 16 + l15], cf[2][v]);
      atomicAdd(&wfg[kr * FC + 3 * 16 + l15], cf[3][v]);
    }
  }
}

// ---------------------------------------------------------------- Kabsch
__device__ inline void kabsch3(const float P[3][3], float R[3][3])
{
  float S[3][3], V[3][3];
  for (int i = 0; i < 3; ++i)
    for (int j = 0; j < 3; ++j) {
      float s = 0.f;
      for (int d = 0; d < 3; ++d) s += P[d][i] * P[d][j];
      S[i][j] = s;
      V[i][j] = (i == j) ? 1.f : 0.f;
    }
  for (int sweep = 0; sweep < 12; ++sweep) {
    for (int pi = 0; pi < 3; ++pi) {
      for (int qi = pi + 1; qi < 3; ++qi) {
        float apq = S[pi][qi];
        if (fabsf(apq) < 1e-22f) continue;
        float tau = (S[qi][qi] - S[pi][pi]) / (2.f * apq);
        float tt  = (tau >= 0.f ? 1.f : -1.f) / (fabsf(tau) + sqrtf(1.f + tau * tau));
        float cc  = 1.f / sqrtf(1.f + tt * tt);
        float ss  = tt * cc;
        for (int r = 0; r < 3; ++r) {
          float a = S[r][pi], b = S[r][qi];
          S[r][pi] = cc * a - ss * b; S[r][qi] = ss * a + cc * b;
        }
        for (int r = 0; r < 3; ++r) {
          float a = S[pi][r], b = S[qi][r];
          S[pi][r] = cc * a - ss * b; S[qi][r] = ss * a + cc * b;
        }
        for (int r = 0; r < 3; ++r) {
          float a = V[r][pi], b = V[r][qi];
          V[r][pi] = cc * a - ss * b; V[r][qi] = ss * a + cc * b;
        }
      }
    }
  }
  float ev[3] = {S[0][0], S[1][1], S[2][2]};
  int idx[3] = {0, 1, 2};
  for (int i = 0; i < 2; ++i)
    for (int j = 0; j < 2 - i; ++j)
      if (ev[idx[j]] < ev[idx[j + 1]]) { int t = idx[j]; idx[j] = idx[j + 1]; idx[j + 1] = t; }
  float v0[3], v1[3], v2[3];
  for (int d = 0; d < 3; ++d) { v0[d] = V[d][idx[0]]; v1[d] = V[d][idx[1]]; v2[d] = V[d][idx[2]]; }
  float cr[3] = {v1[1] * v2[2] - v1[2] * v2[1],
                 v1[2] * v2[0] - v1[0] * v2[2],
                 v1[0] * v2[1] - v1[1] * v2[0]};
  float detV = v0[0] * cr[0] + v0[1] * cr[1] + v0[2] * cr[2];
  float u0[3], u1[3];
  for (int d = 0; d < 3; ++d) {
    u0[d] = P[d][0] * v0[0] + P[d][1] * v0[1] + P[d][2] * v0[2];
    u1[d] = P[d][0] * v1[0] + P[d][1] * v1[1] + P[d][2] * v1[2];
  }
  float n0 = sqrtf(u0[0] * u0[0] + u0[1] * u0[1] + u0[2] * u0[2]);
  if (n0 < 1e-12f) { u0[0] = 1.f; u0[1] = 0.f; u0[2] = 0.f; n0 = 1.f; }
  for (int d = 0; d < 3; ++d) u0[d] /= n0;
  float d01 = u0[0] * u1[0] + u0[1] * u1[1] + u0[2] * u1[2];
  for (int d = 0; d < 3; ++d) u1[d] -= d01 * u0[d];
  float n1 = sqrtf(u1[0] * u1[0] + u1[1] * u1[1] + u1[2] * u1[2]);
  if (n1 < 1e-12f) {
    float wv[3] = {(fabsf(u0[0]) < 0.9f) ? 1.f : 0.f, (fabsf(u0[0]) < 0.9f) ? 0.f : 1.f, 0.f};
    u1[0] = u0[1] * wv[2] - u0[2] * wv[1];
    u1[1] = u0[2] * wv[0] - u0[0] * wv[2];
    u1[2] = u0[0] * wv[1] - u0[1] * wv[0];
    n1 = sqrtf(u1[0] * u1[0] + u1[1] * u1[1] + u1[2] * u1[2]) + 1e-20f;
  }
  for (int d = 0; d < 3; ++d) u1[d] /= n1;
  float u2[3] = {u0[1] * u1[2] - u0[2] * u1[1],
                 u0[2] * u1[0] - u0[0] * u1[2],
                 u0[0] * u1[1] - u0[1] * u1[0]};
  for (int d = 0; d < 3; ++d)
    for (int e = 0; e < 3; ++e)
      R[d][e] = u0[d] * v0[e] + u1[d] * v1[e] + detV * u2[d] * v2[e];
}

// ---------------------------------------------------------------- M-step
__global__ __launch_bounds__(256) void kProc(
    const float* __restrict__ Lg, const float* __restrict__ Wg,
    const float* __restrict__ S2g, const float* __restrict__ wf,
    float* __restrict__ Rg, float* __restrict__ tg, float* __restrict__ Xg,
    float* __restrict__ Qw, float* __restrict__ thT, int iter)
{
  __shared__ float sR[MV][9];
  __shared__ float sT[MV][3];
  const int tid = threadIdx.x;

  if (tid < MV) {
    const int m = tid;
    float mW[3] = {0, 0, 0}, mX[3] = {0, 0, 0}, z = 0.f;
    float PW[3][3] = {{0, 0, 0}, {0, 0, 0}, {0, 0, 0}};
    for (int k = 0; k < KC; ++k) {
      float q  = Qw[k];
      float b  = Lg[m * KC + k] * q;
      float w0 = Wg[(m * 3 + 0) * KC + k] * q;
      float w1 = Wg[(m * 3 + 1) * KC + k] * q;
      float w2 = Wg[(m * 3 + 2) * KC + k] * q;
      float x0 = Xg[k * 3], x1 = Xg[k * 3 + 1], x2 = Xg[k * 3 + 2];
      mW[0] += w0; mW[1] += w1; mW[2] += w2; z += b;
      mX[0] += b * x0; mX[1] += b * x1; mX[2] += b * x2;
      PW[0][0] += w0 * x0; PW[0][1] += w0 * x1; PW[0][2] += w0 * x2;
      PW[1][0] += w1 * x0; PW[1][1] += w1 * x1; PW[1][2] += w1 * x2;
      PW[2][0] += w2 * x0; PW[2][1] += w2 * x1; PW[2][2] += w2 * x2;
    }
    const float zi = 1.f / z;
    float P[3][3];
    for (int d = 0; d < 3; ++d)
      for (int e = 0; e < 3; ++e)
        P[d][e] = PW[e][d] - mX[d] * mW[e] * zi;   // (W@X)^T - mX mW^T / z
    float R[3][3];
    kabsch3(P, R);
    float t[3];
    for (int d = 0; d < 3; ++d)
      t[d] = (mX[d] - (R[d][0] * mW[0] + R[d][1] * mW[1] + R[d][2] * mW[2])) * zi;
    for (int i = 0; i < 9; ++i) { sR[m][i] = R[i / 3][i % 3]; Rg[m * 9 + i] = sR[m][i]; }
    for (int d = 0; d < 3; ++d) { sT[m][d] = t[d]; tg[m * 3 + d] = t[d]; }
  }
  __syncthreads();

  if (tid < KC) {
    const int k = tid;
    float den = 0.f, Xn[3] = {0, 0, 0}, S2 = 0.f;
    for (int m = 0; m < MV; ++m) {
      float Lmk = Lg[m * KC + k];
      float w0 = Wg[(m * 3 + 0) * KC + k];
      float w1 = Wg[(m * 3 + 1) * KC + k];
      float w2 = Wg[(m * 3 + 2) * KC + k];
      den += Lmk;
      for (int d = 0; d < 3; ++d)
        Xn[d] += sR[m][d * 3 + 0] * w0 + sR[m][d * 3 + 1] * w1 +
                 sR[m][d * 3 + 2] * w2 + sT[m][d] * Lmk;
      float rt0 = sR[m][0] * sT[m][0] + sR[m][3] * sT[m][1] + sR[m][6] * sT[m][2];
      float rt1 = sR[m][1] * sT[m][0] + sR[m][4] * sT[m][1] + sR[m][7] * sT[m][2];
      float rt2 = sR[m][2] * sT[m][0] + sR[m][5] * sT[m][1] + sR[m][8] * sT[m][2];
      float tt  = sT[m][0] * sT[m][0] + sT[m][1] * sT[m][1] + sT[m][2] * sT[m][2];
      S2 += S2g[m * KC + k] + 2.f * (rt0 * w0 + rt1 * w1 + rt2 * w2) + tt * Lmk;
    }
    float Xk[3];
    const float di = 1.f / den;
    if (iter > 0) {
      for (int d = 0; d < 3; ++d) { Xk[d] = Xn[d] * di; Xg[k * 3 + d] = Xk[d]; }
    } else {
      for (int d = 0; d < 3; ++d) Xk[d] = Xg[k * 3 + d];
    }
    float wn = S2 - 2.f * (Xk[0] * Xn[0] + Xk[1] * Xn[1] + Xk[2] * Xn[2]) +
               (Xk[0] * Xk[0] + Xk[1] * Xk[1] + Xk[2] * Xk[2]) * den;
    Qw[k] = 3.f * den / (wn + 3.f * den * EPS_C);
    float s = 0.f;
    for (int f = 0; f < FC; ++f) s += wf[k * FC + f];
    const float inv = 1.f / (s + 1e-12f);
    for (int f = 0; f < FC; ++f) thT[f * KC + k] = wf[k * FC + f] * inv;
  }
}

// ---------------------------------------------------------------- launcher
extern "C" void kernel_launch(void* const* d_in, const int* in_sizes, int n_in,
                              void* d_out, int out_size, void* d_ws, size_t ws_size,
                              hipStream_t stream) {
  (void)n_in; (void)out_size; (void)ws_size;
  const float* Vs   = (const float*)d_in[0];
  const float* feat = (const float*)d_in[1];
  const float* X0   = (const float*)d_in[2];
  const float* Q0   = (const float*)d_in[3];
  const float* th0  = (const float*)d_in[4];
  const int N = in_sizes[0] / (MV * 3);

  float* out = (float*)d_out;
  float* Rg  = out;                          // [M,3,3]
  float* tg  = out + MV * 9;                 // [M,3,1]
  float* Xg  = out + MV * 9 + MV * 3;        // [K,3]
  float* TVg = out + MV * 9 + MV * 3 + KC * 3;  // [M,3,N] (also working buffer)

  float* ws   = (float*)d_ws;
  float* Qw    = ws + WS_Q;
  float* betaW = ws + WS_BETA;
  float* Lw    = ws + WS_L;
  float* Ww    = ws + WS_W;
  float* S2w   = ws + WS_S2;
  float* wfW   = ws + WS_WF;
  float* thT   = ws + WS_THT;
  float* acd   = ws + WS_ACD;

  kInit<<<64, 256, 0, stream>>>(X0, Q0, th0, Xg, Qw, betaW, thT);
  kMean<<<MV * 3, 256, 0, stream>>>(Vs, Rg, tg, N);
  dim3 gTV((N + 255) / 256, MV);
  kTV<<<gTV, 256, 0, stream>>>(Vs, Rg, tg, TVg, N);

  for (int it = 0; it < 10; ++it) {
    kZero<<<(21504 + 255) / 256, 256, 0, stream>>>(ws);
    kE<<<dim3((N + 63) / 64, MV), 128, 0, stream>>>(Vs, feat, TVg, Xg, Qw, thT, betaW,
                                                    Lw, Ww, S2w, acd, N);
    kWF<<<dim3((N + 1023) / 1024, MV), 128, 0, stream>>>(feat, TVg, Xg, Qw, acd, wfW, N);
    kProc<<<1, 256, 0, stream>>>(Lw, Ww, S2w, wfW, Rg, tg, Xg, Qw, thT, it);
    kTV<<<gTV, 256, 0, stream>>>(Vs, Rg, tg, TVg, N);
  }
}